// GungnirHalfKA_40209483825328
// MI455X (gfx1250) — compile-verified
//
#include <hip/hip_runtime.h>

// ---------------------------------------------------------------------------
// Fused NNUE forward for MI455X (gfx1250, wave32).
//   Pre-pass (if ws fits): fake-quant ft table f32 -> int16 (halves L2 traffic,
//                          92MB table is L2-resident), fake-quant fc0_w -> f32.
//   Phase 1: sparse gather-sum with packed i16 adds + pairwise clip -> ft in LDS
//   Phase 2: fc0 for all 8 stacks via v_wmma_f32_16x16x4_f32 (f32-exact)
//   Phase 3: bucket select + fc1/fc2 tail + psqt, one lane per sample
// ---------------------------------------------------------------------------

typedef __attribute__((ext_vector_type(2))) float v2f;
typedef __attribute__((ext_vector_type(8))) float v8f;
typedef __attribute__((ext_vector_type(2))) short s16x2;

#define FT_OUT   1024
#define NSTACK   8
#define FC0_OUT  16
#define TILE     16
#define TPB      2        // tiles per block
#define PITCH    1028     // ft row pitch in floats (bank-conflict padding)

__device__ __forceinline__ float q16f(float x) {
  return fminf(fmaxf(rintf(x), -32768.f), 32767.f);
}
__device__ __forceinline__ float q8f(float x) {
  return fminf(fmaxf(rintf(x), -128.f), 127.f);
}
__device__ __forceinline__ float q32f(float x) {
  return fminf(fmaxf(rintf(x), -2147483648.f), 2147483647.f);
}
__device__ __forceinline__ float clip127(float x) {
  return fminf(fmaxf(x, 0.f), 127.f);
}

// ---- pre-pass: fake-quant ft table to int16 -------------------------------
__global__ __launch_bounds__(256) void quant_ft_kernel(
    const float* __restrict__ ftw, short* __restrict__ dst, int n4) {
  int i = blockIdx.x * 256 + threadIdx.x;
  if (i < n4) {
    float4 v = ((const float4*)ftw)[i];
    short4 q;
    q.x = (short)q16f(v.x);
    q.y = (short)q16f(v.y);
    q.z = (short)q16f(v.z);
    q.w = (short)q16f(v.w);
    ((short4*)dst)[i] = q;
  }
}

// ---- pre-pass: fake-quant fc0 weights to f32 ------------------------------
__global__ __launch_bounds__(256) void quant_fc0_kernel(
    const float* __restrict__ src, float* __restrict__ dst, int n) {
  int i = blockIdx.x * 256 + threadIdx.x;
  if (i < n) dst[i] = q8f(src[i]);
}

// ---------------------------------------------------------------------------
template <bool PREQ>
__global__ __launch_bounds__(128) void nnue_fused_kernel(
    const int* __restrict__ wfeat, const int* __restrict__ woff,
    const int* __restrict__ bfeat, const int* __restrict__ boff,
    const int* __restrict__ stm,   const int* __restrict__ bucket,
    const float* __restrict__ ftw, const short* __restrict__ ftwq,
    const float* __restrict__ ftb,
    const float* __restrict__ psq,
    const float* __restrict__ fc0w,   // PREQ ? pre-quantized f32 : raw
    const float* __restrict__ fc0b,
    const float* __restrict__ fc1w, const float* __restrict__ fc1b,
    const float* __restrict__ fc2w, const float* __restrict__ fc2b,
    float* __restrict__ out, int B, int nTiles)
{
  __shared__ float ft_lds[TILE * PITCH];              // 16 x 1024 ft tile (padded)
  __shared__ float o0_lds[NSTACK][TILE][FC0_OUT];     // fc0 outputs, all stacks
  __shared__ int   feats_lds[TILE][64];               // 32 w-feats + 32 b-feats

  const int tid  = threadIdx.x;
  const int lane = tid & 31;
  const int wv   = tid >> 5;
  const int r    = lane & 15;         // M/N index inside wmma fragments
  const int hi   = (lane >> 4) & 1;   // half-wave selector

  for (int t = 0; t < TPB; ++t) {
    int tile = blockIdx.x * TPB + t;
    if (tile >= nTiles) break;

    // ---- stage feature indices for this tile ----
    for (int k = tid; k < TILE * 64; k += 128) {
      int row = k >> 6, idx = k & 63;
      int s = tile * TILE + row;
      int v = 0;
      if (s < B) v = (idx < 32) ? wfeat[woff[s] + idx]
                                : bfeat[boff[s] + (idx - 32)];
      feats_lds[row][idx] = v;
    }
    __syncthreads();

    // =====================================================================
    // Phase 1: gather-sum + bias + stm-select + pairwise -> ft_lds
    // =====================================================================
    for (int ss = 0; ss < 4; ++ss) {
      int row = wv * 4 + ss;
      int s = tile * TILE + row;

      if (PREQ) {
        // ---- int16 path: lane owns cols {m*256 + 8*lane .. +7}, m=0..3.
        // Pair (h, h+512): chunk m pairs with chunk m+2 -> lane-local.
        s16x2 accW[16], accB[16];
        #pragma unroll
        for (int k = 0; k < 16; ++k) {
          s16x2 z = {0, 0};
          accW[k] = z; accB[k] = z;
        }
        if (s < B) {
          for (int i = 0; i < 32; ++i) {
            const int4* p = (const int4*)(ftwq + (size_t)feats_lds[row][i] * FT_OUT);
            #pragma unroll
            for (int m = 0; m < 4; ++m) {
              int4 v = p[m * 32 + lane];
              accW[m*4+0] += __builtin_bit_cast(s16x2, v.x);
              accW[m*4+1] += __builtin_bit_cast(s16x2, v.y);
              accW[m*4+2] += __builtin_bit_cast(s16x2, v.z);
              accW[m*4+3] += __builtin_bit_cast(s16x2, v.w);
            }
          }
          for (int i = 0; i < 32; ++i) {
            const int4* p = (const int4*)(ftwq + (size_t)feats_lds[row][32 + i] * FT_OUT);
            #pragma unroll
            for (int m = 0; m < 4; ++m) {
              int4 v = p[m * 32 + lane];
              accB[m*4+0] += __builtin_bit_cast(s16x2, v.x);
              accB[m*4+1] += __builtin_bit_cast(s16x2, v.y);
              accB[m*4+2] += __builtin_bit_cast(s16x2, v.z);
              accB[m*4+3] += __builtin_bit_cast(s16x2, v.w);
            }
          }
          // unpack to float, add q16 bias
          float fW[32], fB[32];
          #pragma unroll
          for (int m = 0; m < 4; ++m) {
            float4 b0 = ((const float4*)ftb)[m * 64 + 2 * lane];
            float4 b1 = ((const float4*)ftb)[m * 64 + 2 * lane + 1];
            float qb[8] = { q16f(b0.x), q16f(b0.y), q16f(b0.z), q16f(b0.w),
                            q16f(b1.x), q16f(b1.y), q16f(b1.z), q16f(b1.w) };
            #pragma unroll
            for (int d = 0; d < 4; ++d) {
              fW[m*8 + d*2 + 0] = (float)(int)accW[m*4+d].x + qb[d*2+0];
              fW[m*8 + d*2 + 1] = (float)(int)accW[m*4+d].y + qb[d*2+1];
              fB[m*8 + d*2 + 0] = (float)(int)accB[m*4+d].x + qb[d*2+0];
              fB[m*8 + d*2 + 1] = (float)(int)accB[m*4+d].y + qb[d*2+1];
            }
          }
          int st = stm[s];   // wave-uniform
          if (st) {
            #pragma unroll
            for (int k = 0; k < 32; ++k) { float tv = fW[k]; fW[k] = fB[k]; fB[k] = tv; }
          }
          // pairwise -> ft_lds
          #pragma unroll
          for (int m = 0; m < 2; ++m) {
            float4 x0, x1, y0, y1;
            x0.x = clip127(fW[m*8+0]) * clip127(fW[(m+2)*8+0]) * (1.f/128.f);
            x0.y = clip127(fW[m*8+1]) * clip127(fW[(m+2)*8+1]) * (1.f/128.f);
            x0.z = clip127(fW[m*8+2]) * clip127(fW[(m+2)*8+2]) * (1.f/128.f);
            x0.w = clip127(fW[m*8+3]) * clip127(fW[(m+2)*8+3]) * (1.f/128.f);
            x1.x = clip127(fW[m*8+4]) * clip127(fW[(m+2)*8+4]) * (1.f/128.f);
            x1.y = clip127(fW[m*8+5]) * clip127(fW[(m+2)*8+5]) * (1.f/128.f);
            x1.z = clip127(fW[m*8+6]) * clip127(fW[(m+2)*8+6]) * (1.f/128.f);
            x1.w = clip127(fW[m*8+7]) * clip127(fW[(m+2)*8+7]) * (1.f/128.f);
            y0.x = clip127(fB[m*8+0]) * clip127(fB[(m+2)*8+0]) * (1.f/128.f);
            y0.y = clip127(fB[m*8+1]) * clip127(fB[(m+2)*8+1]) * (1.f/128.f);
            y0.z = clip127(fB[m*8+2]) * clip127(fB[(m+2)*8+2]) * (1.f/128.f);
            y0.w = clip127(fB[m*8+3]) * clip127(fB[(m+2)*8+3]) * (1.f/128.f);
            y1.x = clip127(fB[m*8+4]) * clip127(fB[(m+2)*8+4]) * (1.f/128.f);
            y1.y = clip127(fB[m*8+5]) * clip127(fB[(m+2)*8+5]) * (1.f/128.f);
            y1.z = clip127(fB[m*8+6]) * clip127(fB[(m+2)*8+6]) * (1.f/128.f);
            y1.w = clip127(fB[m*8+7]) * clip127(fB[(m+2)*8+7]) * (1.f/128.f);
            float* base = &ft_lds[row * PITCH + m * 256 + 8 * lane];
            *(float4*)(base + 0)       = x0;
            *(float4*)(base + 4)       = x1;
            *(float4*)(base + 512 + 0) = y0;
            *(float4*)(base + 512 + 4) = y1;
          }
        } else {
          float4 z; z.x = z.y = z.z = z.w = 0.f;
          #pragma unroll
          for (int m = 0; m < 2; ++m) {
            float* base = &ft_lds[row * PITCH + m * 256 + 8 * lane];
            *(float4*)(base + 0)       = z;
            *(float4*)(base + 4)       = z;
            *(float4*)(base + 512 + 0) = z;
            *(float4*)(base + 512 + 4) = z;
          }
        }
      } else {
        // ---- fallback f32 path with inline fake-quant (round-1 design):
        // lane owns cols {j*128 + 4*lane .. +3}, j = 0..7.
        float aW[32], aB[32];
        #pragma unroll
        for (int k = 0; k < 32; ++k) { aW[k] = 0.f; aB[k] = 0.f; }
        if (s < B) {
          for (int i = 0; i < 32; ++i) {
            const float4* p = (const float4*)(ftw + (size_t)feats_lds[row][i] * FT_OUT);
            #pragma unroll
            for (int j = 0; j < 8; ++j) {
              float4 v = p[j * 32 + lane];
              aW[j*4+0] += q16f(v.x); aW[j*4+1] += q16f(v.y);
              aW[j*4+2] += q16f(v.z); aW[j*4+3] += q16f(v.w);
            }
          }
          for (int i = 0; i < 32; ++i) {
            const float4* p = (const float4*)(ftw + (size_t)feats_lds[row][32 + i] * FT_OUT);
            #pragma unroll
            for (int j = 0; j < 8; ++j) {
              float4 v = p[j * 32 + lane];
              aB[j*4+0] += q16f(v.x); aB[j*4+1] += q16f(v.y);
              aB[j*4+2] += q16f(v.z); aB[j*4+3] += q16f(v.w);
            }
          }
          #pragma unroll
          for (int j = 0; j < 8; ++j) {
            float4 v = ((const float4*)ftb)[j * 32 + lane];
            float b0 = q16f(v.x), b1 = q16f(v.y), b2 = q16f(v.z), b3 = q16f(v.w);
            aW[j*4+0] += b0; aW[j*4+1] += b1; aW[j*4+2] += b2; aW[j*4+3] += b3;
            aB[j*4+0] += b0; aB[j*4+1] += b1; aB[j*4+2] += b2; aB[j*4+3] += b3;
          }
          int st = stm[s];
          if (st) {
            #pragma unroll
            for (int k = 0; k < 32; ++k) { float tv = aW[k]; aW[k] = aB[k]; aB[k] = tv; }
          }
          #pragma unroll
          for (int j = 0; j < 4; ++j) {
            float4 x, y;
            x.x = clip127(aW[j*4+0]) * clip127(aW[(j+4)*4+0]) * (1.f/128.f);
            x.y = clip127(aW[j*4+1]) * clip127(aW[(j+4)*4+1]) * (1.f/128.f);
            x.z = clip127(aW[j*4+2]) * clip127(aW[(j+4)*4+2]) * (1.f/128.f);
            x.w = clip127(aW[j*4+3]) * clip127(aW[(j+4)*4+3]) * (1.f/128.f);
            y.x = clip127(aB[j*4+0]) * clip127(aB[(j+4)*4+0]) * (1.f/128.f);
            y.y = clip127(aB[j*4+1]) * clip127(aB[(j+4)*4+1]) * (1.f/128.f);
            y.z = clip127(aB[j*4+2]) * clip127(aB[(j+4)*4+2]) * (1.f/128.f);
            y.w = clip127(aB[j*4+3]) * clip127(aB[(j+4)*4+3]) * (1.f/128.f);
            *(float4*)&ft_lds[row * PITCH + j * 128 + lane * 4]       = x;
            *(float4*)&ft_lds[row * PITCH + 512 + j * 128 + lane * 4] = y;
          }
        } else {
          float4 z; z.x = z.y = z.z = z.w = 0.f;
          #pragma unroll
          for (int j = 0; j < 4; ++j) {
            *(float4*)&ft_lds[row * PITCH + j * 128 + lane * 4]       = z;
            *(float4*)&ft_lds[row * PITCH + 512 + j * 128 + lane * 4] = z;
          }
        }
      }
    }
    __syncthreads();

    // =====================================================================
    // Phase 2: fc0 for all 8 stacks, WMMA f32 16x16x4, K = 1024.
    // =====================================================================
    {
      int s0 = wv * 2, s1 = wv * 2 + 1;
      const float* Bp0 = fc0w + ((size_t)s0 * FC0_OUT + r) * FT_OUT;
      const float* Bp1 = fc0w + ((size_t)s1 * FC0_OUT + r) * FT_OUT;
      const float* Ap  = &ft_lds[r * PITCH];
      v8f acc0 = {0.f,0.f,0.f,0.f,0.f,0.f,0.f,0.f};
      v8f acc1 = {0.f,0.f,0.f,0.f,0.f,0.f,0.f,0.f};
      #pragma unroll 4
      for (int kk = 0; kk < 256; ++kk) {
        int c = kk * 4 + hi * 2;
        v2f a  = *(const v2f*)(Ap + c);       // ds_load_b64
        v2f b0 = *(const v2f*)(Bp0 + c);      // global b64 (L2-hot 512KB table)
        v2f b1 = *(const v2f*)(Bp1 + c);
        if (!PREQ) {
          b0.x = q8f(b0.x); b0.y = q8f(b0.y);
          b1.x = q8f(b1.x); b1.y = q8f(b1.y);
        }
        acc0 = __builtin_amdgcn_wmma_f32_16x16x4_f32(
            false, a, false, b0, (short)0, acc0, false, false);
        acc1 = __builtin_amdgcn_wmma_f32_16x16x4_f32(
            false, a, false, b1, (short)0, acc1, false, false);
      }
      #pragma unroll
      for (int v = 0; v < 8; ++v) {
        o0_lds[s0][v + hi * 8][r] = acc0[v];
        o0_lds[s1][v + hi * 8][r] = acc1[v];
      }
    }
    __syncthreads();

    // =====================================================================
    // Phase 3: per-sample tail (bucket select, fc1, fc2, skip, psqt)
    // =====================================================================
    if (wv == 0 && lane < 16) {
      int s = tile * TILE + lane;
      if (s < B) {
        int bk = bucket[s];
        int st = stm[s];
        float o0[16];
        #pragma unroll
        for (int o = 0; o < 16; ++o)
          o0[o] = o0_lds[bk][lane][o] + q32f(fc0b[bk * FC0_OUT + o]);

        float slab[32];
        #pragma unroll
        for (int i = 0; i < 15; ++i) {
          slab[i]      = clip127(o0[i] * o0[i] * (1.f / 524288.f));
          slab[15 + i] = clip127(o0[i] * (1.f / 64.f));
        }
        slab[30] = 0.f; slab[31] = 0.f;

        float ac1[32];
        const float* w1 = fc1w + (size_t)bk * 32 * 32;
        #pragma unroll 2
        for (int o = 0; o < 32; ++o) {
          float acc = q32f(fc1b[bk * 32 + o]);
          for (int i = 0; i < 30; ++i)
            acc += slab[i] * q8f(w1[o * 32 + i]);
          ac1[o] = clip127(acc * (1.f / 64.f));
        }

        float scalar = q32f(fc2b[bk]);
        const float* w2 = fc2w + (size_t)bk * 32;
        for (int i = 0; i < 32; ++i)
          scalar += ac1[i] * q8f(w2[i]);

        float skip = o0[15] * (9600.f / 8128.f);

        float pw = 0.f, pb = 0.f;
        for (int i = 0; i < 32; ++i)
          pw += q32f(psq[(size_t)feats_lds[lane][i] * NSTACK + bk]);
        for (int i = 0; i < 32; ++i)
          pb += q32f(psq[(size_t)feats_lds[lane][32 + i] * NSTACK + bk]);
        float ps = st ? pb : pw;
        float po = st ? pw : pb;
        float psqt_v = (ps - po) * 0.5f;

        out[s] = (psqt_v + scalar + skip) * (1.f / 16.f);
      }
    }
    __syncthreads();
  }
}

extern "C" void kernel_launch(void* const* d_in, const int* in_sizes, int n_in,
                              void* d_out, int out_size, void* d_ws, size_t ws_size,
                              hipStream_t stream) {
  const int*   wfeat = (const int*)d_in[0];
  const int*   woff  = (const int*)d_in[1];
  const int*   bfeat = (const int*)d_in[2];
  const int*   boff  = (const int*)d_in[3];
  const int*   stm   = (const int*)d_in[4];
  const int*   buck  = (const int*)d_in[5];
  const float* ftw   = (const float*)d_in[6];
  const float* ftb   = (const float*)d_in[7];
  const float* psq   = (const float*)d_in[8];
  const float* fc0w  = (const float*)d_in[9];
  const float* fc0b  = (const float*)d_in[10];
  const float* fc1w  = (const float*)d_in[11];
  const float* fc1b  = (const float*)d_in[12];
  const float* fc2w  = (const float*)d_in[13];
  const float* fc2b  = (const float*)d_in[14];
  float* out = (float*)d_out;

  int B = in_sizes[4];                    // stm has one entry per sample
  int nTiles = (B + TILE - 1) / TILE;
  int nBlocks = (nTiles + TPB - 1) / TPB;

  const int ftElems  = in_sizes[6];       // 45056 * 1024
  const int fc0Elems = in_sizes[9];       // 8 * 16 * 1024
  size_t wsNeed = (size_t)ftElems * sizeof(short) + (size_t)fc0Elems * sizeof(float);

  if (ws_size >= wsNeed) {
    short* ftwq = (short*)d_ws;
    float* fc0q = (float*)((char*)d_ws + (size_t)ftElems * sizeof(short));

    int n4 = ftElems / 4;
    quant_ft_kernel<<<dim3((n4 + 255) / 256), dim3(256), 0, stream>>>(ftw, ftwq, n4);
    quant_fc0_kernel<<<dim3((fc0Elems + 255) / 256), dim3(256), 0, stream>>>(fc0w, fc0q, fc0Elems);

    nnue_fused_kernel<true><<<dim3(nBlocks), dim3(128), 0, stream>>>(
        wfeat, woff, bfeat, boff, stm, buck,
        ftw, (const short*)ftwq, ftb, psq,
        (const float*)fc0q, fc0b, fc1w, fc1b, fc2w, fc2b,
        out, B, nTiles);
  } else {
    nnue_fused_kernel<false><<<dim3(nBlocks), dim3(128), 0, stream>>>(
        wfeat, woff, bfeat, boff, stm, buck,
        ftw, (const short*)nullptr, ftb, psq,
        fc0w, fc0b, fc1w, fc1b, fc2w, fc2b,
        out, B, nTiles);
  }
}